// AttentionLayer_29678224016048
// MI455X (gfx1250) — compile-verified
//
#include <hip/hip_runtime.h>

// MI455X single-pass online-softmax attention pooling.
// HBM-bound: 512MB of x read exactly once (~22us floor at 23.3 TB/s).
// Scores via V_WMMA_F32_16X16X4_F32: A = W replicated (so C[:,n] = score_n,
// every lane ends up holding its row's score directly, no cross-lane dot).

typedef __attribute__((ext_vector_type(2))) float v2f;
typedef __attribute__((ext_vector_type(4))) float v4f;
typedef __attribute__((ext_vector_type(8))) float v8f;

#define GRID 512
#define BLOCK 256
#define WAVES_PER_BLOCK (BLOCK / 32)
#define TOTAL_WAVES (GRID * WAVES_PER_BLOCK)   // 4096 partials (~2MB ws)
#define DIM 128

__device__ __forceinline__ float readlane_f(float v, int l) {
  return __builtin_bit_cast(float,
      __builtin_amdgcn_readlane(__builtin_bit_cast(int, v), l));
}

__global__ __launch_bounds__(BLOCK) void attn_pass1(
    const float* __restrict__ x, const float* __restrict__ W,
    float* __restrict__ wsAcc, float* __restrict__ wsStats, int n)
{
  const int lane = threadIdx.x & 31;
  const int h    = lane >> 4;          // 0 or 1 (K half within A/B layout)
  const int lr   = lane & 15;          // row-in-tile this lane owns in B/C
  const int wave = blockIdx.x * (blockDim.x >> 5) + (threadIdx.x >> 5);
  const int numTiles = n >> 4;

  // Preload A (= W broadcast into all 16 rows) in the 16x4 f32 A-layout for
  // all 32 K-steps. A-layout: lanes 0-15 need {W[4k],W[4k+1]}, lanes 16-31
  // need {W[4k+2],W[4k+3]}  ->  b64 at byte offset 16k + 8h.
  v2f A[32];
  {
    const v2f* wp = (const v2f*)(W + 2 * h);
    #pragma unroll
    for (int kk = 0; kk < 32; ++kk) A[kk] = wp[2 * kk];
  }

  float m = -__builtin_inff();
  float Z = 0.0f;
  v4f acc = {0.0f, 0.0f, 0.0f, 0.0f};

  for (int t = wave; t < numTiles; t += TOTAL_WAVES) {
    const int r0 = t << 4;

    // ---- scores for 16 rows via 32x V_WMMA_F32_16X16X4_F32 ----
    // B-layout (4x16): lane n holds x[r0+n%16][4k + 2h + {0,1}] -> b64 loads.
    const float* bp = x + (size_t)(r0 + lr) * DIM + 2 * h;
    v8f c = {};
    #pragma unroll
    for (int kk = 0; kk < 32; ++kk) {
      v2f bv = *(const v2f*)(bp + 4 * kk);
      c = __builtin_amdgcn_wmma_f32_16x16x4_f32(
          false, A[kk], false, bv, (short)0, c, false, false);
    }
    // All A rows equal W => every C reg in lane holds score of row (lane%16).
    float s = c[0];

    // tile max (duplicated lanes don't affect max)
    float tm = s;
    tm = fmaxf(tm, __shfl_xor(tm, 16, 32));
    tm = fmaxf(tm, __shfl_xor(tm,  8, 32));
    tm = fmaxf(tm, __shfl_xor(tm,  4, 32));
    tm = fmaxf(tm, __shfl_xor(tm,  2, 32));
    tm = fmaxf(tm, __shfl_xor(tm,  1, 32));

    float mnew  = fmaxf(m, tm);
    float scale = __expf(m - mnew);        // first tile: exp(-inf)=0
    float p     = __expf(s - mnew);

    // sum of 16 distinct p (each appears twice across 32 lanes)
    float ps = p;
    ps += __shfl_xor(ps, 16, 32);
    ps += __shfl_xor(ps,  8, 32);
    ps += __shfl_xor(ps,  4, 32);
    ps += __shfl_xor(ps,  2, 32);
    ps += __shfl_xor(ps,  1, 32);
    Z = Z * scale + 0.5f * ps;

    acc = acc * scale;

    // ---- weighted accumulation: lane owns dims [4*lane, 4*lane+4) ----
    // Coalesced float4 reloads hit WGP$ (tile was just streamed in).
    const float* xr = x + (size_t)r0 * DIM + 4 * lane;
    #pragma unroll
    for (int i = 0; i < 16; ++i) {
      float sp = readlane_f(p, i);                 // uniform broadcast
      v4f xv = *(const v4f*)(xr + i * DIM);
      acc += sp * xv;
    }
    m = mnew;
  }

  // tail rows (n % 16 != 0) — dead for N=1e6, kept for safety
  const int tailStart = numTiles << 4;
  if ((tailStart < n) && (wave == TOTAL_WAVES - 1)) {
    v4f wv = *(const v4f*)(W + 4 * lane);
    for (int r = tailStart; r < n; ++r) {
      v4f xv = *(const v4f*)(x + (size_t)r * DIM + 4 * lane);
      float d = xv.x * wv.x + xv.y * wv.y + xv.z * wv.z + xv.w * wv.w;
      d += __shfl_xor(d, 16, 32);
      d += __shfl_xor(d,  8, 32);
      d += __shfl_xor(d,  4, 32);
      d += __shfl_xor(d,  2, 32);
      d += __shfl_xor(d,  1, 32);
      float mnew  = fmaxf(m, d);
      float scale = __expf(m - mnew);
      float p     = __expf(d - mnew);
      Z   = Z * scale + p;
      acc = acc * scale + p * xv;
      m = mnew;
    }
  }

  // per-wave partial: 128-float acc + (m, Z)
  *((v4f*)(wsAcc + (size_t)wave * DIM) + lane) = acc;
  if (lane == 0) {
    wsStats[2 * wave + 0] = m;
    wsStats[2 * wave + 1] = Z;
  }
}

__global__ __launch_bounds__(128) void attn_pass2(
    const float* __restrict__ wsAcc, const float* __restrict__ wsStats,
    float* __restrict__ out)
{
  const int j = threadIdx.x;   // output dim 0..127
  float M = -__builtin_inff();
  for (int w = 0; w < TOTAL_WAVES; ++w) M = fmaxf(M, wsStats[2 * w]);
  float num = 0.0f, den = 0.0f;
  for (int w = 0; w < TOTAL_WAVES; ++w) {
    float e = __expf(wsStats[2 * w] - M);
    den += e * wsStats[2 * w + 1];
    num += e * wsAcc[(size_t)w * DIM + j];
  }
  out[j] = num / den;
}

extern "C" void kernel_launch(void* const* d_in, const int* in_sizes, int n_in,
                              void* d_out, int out_size, void* d_ws, size_t ws_size,
                              hipStream_t stream) {
  const float* x = (const float*)d_in[0];   // [N,128] f32
  const float* W = (const float*)d_in[1];   // [1,128] f32 (bias cancels in softmax)
  const int n = in_sizes[0] / DIM;

  float* wsAcc   = (float*)d_ws;                               // [TOTAL_WAVES][128]
  float* wsStats = wsAcc + (size_t)TOTAL_WAVES * DIM;          // [TOTAL_WAVES][2]

  attn_pass1<<<GRID, BLOCK, 0, stream>>>(x, W, wsAcc, wsStats, n);
  attn_pass2<<<1, 128, 0, stream>>>(wsAcc, wsStats, (float*)d_out);
}